// RSPR_88648124990728
// MI455X (gfx1250) — compile-verified
//
#include <hip/hip_runtime.h>

typedef float v2f __attribute__((ext_vector_type(2)));
typedef float v8f __attribute__((ext_vector_type(8)));

#define NNODES  50000
#define NEDGES  800000
#define HID     96
#define NGRAPHS 512
#define CLASSES 16

// ---------------------------------------------------------------- utilities
__global__ void RSPR_zero_kernel(float* __restrict__ p, int n) {
    int i = blockIdx.x * blockDim.x + threadIdx.x;
    if (i < n) p[i] = 0.0f;
}

// Per-relation in-degree counts (same for both layers)
__global__ void RSPR_count_kernel(const int* __restrict__ dst,
                                  const int* __restrict__ etype,
                                  float* __restrict__ cnt) {
    int e = blockIdx.x * blockDim.x + threadIdx.x;
    if (e < NEDGES) atomicAdd(&cnt[(size_t)etype[e] * NNODES + dst[e]], 1.0f);
}

__global__ void RSPR_inv_kernel(float* __restrict__ cnt) {
    int i = blockIdx.x * blockDim.x + threadIdx.x;
    if (i < 3 * NNODES) cnt[i] = 1.0f / fmaxf(cnt[i], 1.0f);
}

// ------------------------------------------------------- fused RGCN GEMM
// For a 16-row strip of X: H[r] = X @ Wrel[r] (r=0..2), Y = X @ Wroot + b.
// One wave computes one 16x16 output tile via 24x V_WMMA_F32_16X16X4_F32.
__global__ __launch_bounds__(256)
void RSPR_gemm_kernel(const float* __restrict__ X,
                      const float* __restrict__ Wrel,   // [3,96,96]
                      const float* __restrict__ Wroot,  // [96,96]
                      const float* __restrict__ bias,   // [96]
                      float* __restrict__ H,            // [3,N,96]
                      float* __restrict__ Y,            // [N,96]
                      int relu_in)
{
    __shared__ float Xs[16][100];                       // pad 96->100: conflict-free cols
    const int row0 = blockIdx.x * 16;                   // 50000 = 3125*16, always full

    for (int i = threadIdx.x; i < 16 * HID; i += 256) {
        int r = i / HID, c = i % HID;
        float v = X[(size_t)(row0 + r) * HID + c];
        Xs[r][c] = relu_in ? fmaxf(v, 0.0f) : v;
    }
    __syncthreads();

    const int lane = threadIdx.x & 31;
    const int wave = threadIdx.x >> 5;
    const int half = lane >> 4;      // 0: lanes 0-15, 1: lanes 16-31
    const int l16  = lane & 15;

    // 24 tile-tasks: matrix m = task/6 (0..2 rel, 3 root), col tile = task%6
    for (int task = wave; task < 24; task += 8) {
        const int m    = task / 6;
        const int col0 = (task % 6) * 16;
        const float* __restrict__ W = (m < 3) ? (Wrel + (size_t)m * HID * HID) : Wroot;

        v8f acc = {0.f, 0.f, 0.f, 0.f, 0.f, 0.f, 0.f, 0.f};
#pragma unroll
        for (int k0 = 0; k0 < HID; k0 += 4) {
            const int kk = k0 + 2 * half;   // ISA layout: VGPR0=K0/K2, VGPR1=K1/K3
            v2f a, b;
            a.x = Xs[l16][kk];
            a.y = Xs[l16][kk + 1];
            b.x = W[(size_t)kk * HID + col0 + l16];
            b.y = W[(size_t)(kk + 1) * HID + col0 + l16];
            acc = __builtin_amdgcn_wmma_f32_16x16x4_f32(
                      false, a, false, b, (short)0, acc, false, false);
        }

        // C/D layout: VGPR v holds M=v (lanes 0-15) and M=v+8 (lanes 16-31), N=lane&15
        if (m < 3) {
            float* out = H + ((size_t)m * NNODES + row0) * HID;
#pragma unroll
            for (int v = 0; v < 8; ++v)
                out[(size_t)(v + 8 * half) * HID + col0 + l16] = acc[v];
        } else {
            const float bb = bias[col0 + l16];
            float* out = Y + (size_t)row0 * HID;
#pragma unroll
            for (int v = 0; v < 8; ++v)
                out[(size_t)(v + 8 * half) * HID + col0 + l16] = acc[v] + bb;
        }
    }
}

// ------------------------------------------------- edge scatter (mean aggr)
// One wave per edge: Y[dst] += H[etype][src] * inv_cnt[etype][dst]
__global__ __launch_bounds__(256)
void RSPR_scatter_kernel(const float* __restrict__ H,
                         const float* __restrict__ inv,
                         const int* __restrict__ src,
                         const int* __restrict__ dst,
                         const int* __restrict__ etype,
                         float* __restrict__ Y)
{
    const int e    = (int)((blockIdx.x * blockDim.x + threadIdx.x) >> 5);
    const int lane = threadIdx.x & 31;
    if (e >= NEDGES) return;
    const int s = src[e], d = dst[e], r = etype[e];
    const float sc = inv[(size_t)r * NNODES + d];
    const float* __restrict__ hp = H + ((size_t)r * NNODES + s) * HID;
    float* __restrict__ yp = Y + (size_t)d * HID;
    atomicAdd(&yp[lane],      hp[lane]      * sc);
    atomicAdd(&yp[lane + 32], hp[lane + 32] * sc);
    atomicAdd(&yp[lane + 64], hp[lane + 64] * sc);
}

// ------------------------------------------------------ global mean pool
// One wave per node; ReLU fused on read of layer-2 output.
__global__ __launch_bounds__(256)
void RSPR_pool_kernel(const float* __restrict__ Y,
                      const int* __restrict__ batch,
                      float* __restrict__ pooled,   // [G,96]
                      float* __restrict__ pcnt)     // [G]
{
    const int node = (int)((blockIdx.x * blockDim.x + threadIdx.x) >> 5);
    const int lane = threadIdx.x & 31;
    if (node >= NNODES) return;
    const int g = batch[node];
    const float* __restrict__ yp = Y + (size_t)node * HID;
    float* __restrict__ pp = pooled + (size_t)g * HID;
    atomicAdd(&pp[lane],      fmaxf(yp[lane],      0.0f));
    atomicAdd(&pp[lane + 32], fmaxf(yp[lane + 32], 0.0f));
    atomicAdd(&pp[lane + 64], fmaxf(yp[lane + 64], 0.0f));
    if (lane == 0) atomicAdd(&pcnt[g], 1.0f);
}

// ---------------------------------------------------- final tiny linear
__global__ void RSPR_final_kernel(const float* __restrict__ pooled,
                                  const float* __restrict__ pcnt,
                                  const float* __restrict__ Wlin,  // [96,16]
                                  const float* __restrict__ blin,  // [16]
                                  float* __restrict__ out)         // [512,16]
{
    int i = blockIdx.x * blockDim.x + threadIdx.x;
    if (i >= NGRAPHS * CLASSES) return;
    int g = i / CLASSES, c = i % CLASSES;
    float s = 0.0f;
#pragma unroll 8
    for (int k = 0; k < HID; ++k)
        s += pooled[(size_t)g * HID + k] * Wlin[(size_t)k * CLASSES + c];
    out[i] = s / fmaxf(pcnt[g], 1.0f) + blin[c];
}

// ---------------------------------------------------------------- launch
extern "C" void kernel_launch(void* const* d_in, const int* in_sizes, int n_in,
                              void* d_out, int out_size, void* d_ws, size_t ws_size,
                              hipStream_t stream) {
    const float* x      = (const float*)d_in[0];
    const int*   ei     = (const int*)  d_in[1];   // [2, E]
    const int*   etype  = (const int*)  d_in[2];
    const int*   batch  = (const int*)  d_in[3];
    const float* Wrel1  = (const float*)d_in[4];
    const float* Wroot1 = (const float*)d_in[5];
    const float* b1     = (const float*)d_in[6];
    const float* Wrel2  = (const float*)d_in[7];
    const float* Wroot2 = (const float*)d_in[8];
    const float* b2     = (const float*)d_in[9];
    const float* Wlin   = (const float*)d_in[10];
    const float* blin   = (const float*)d_in[11];
    float* out = (float*)d_out;

    const int* src = ei;
    const int* dst = ei + NEDGES;

    // workspace layout (floats)
    float* ws     = (float*)d_ws;
    float* cnt    = ws;                                   // 3*N      (becomes inv)
    float* H      = cnt + (size_t)3 * NNODES;             // 3*N*96
    float* Y1     = H   + (size_t)3 * NNODES * HID;       // N*96
    float* Y2     = Y1  + (size_t)NNODES * HID;           // N*96
    float* pooled = Y2  + (size_t)NNODES * HID;           // G*96
    float* pcnt   = pooled + (size_t)NGRAPHS * HID;       // G

    const int ZB = 256;
    // degree counts -> reciprocals (shared by both layers)
    RSPR_zero_kernel<<<(3 * NNODES + ZB - 1) / ZB, ZB, 0, stream>>>(cnt, 3 * NNODES);
    RSPR_count_kernel<<<(NEDGES + ZB - 1) / ZB, ZB, 0, stream>>>(dst, etype, cnt);
    RSPR_inv_kernel<<<(3 * NNODES + ZB - 1) / ZB, ZB, 0, stream>>>(cnt);

    const int gemmBlocks    = NNODES / 16;                // 3125, exact
    const int scatterBlocks = (NEDGES * 32) / 256;        // 1 wave / edge
    const int poolBlocks    = (NNODES * 32 + 255) / 256;  // 1 wave / node

    // layer 1
    RSPR_gemm_kernel<<<gemmBlocks, 256, 0, stream>>>(x, Wrel1, Wroot1, b1, H, Y1, 0);
    RSPR_scatter_kernel<<<scatterBlocks, 256, 0, stream>>>(H, cnt, src, dst, etype, Y1);
    // layer 2 (ReLU fused into GEMM input staging)
    RSPR_gemm_kernel<<<gemmBlocks, 256, 0, stream>>>(Y1, Wrel2, Wroot2, b2, H, Y2, 1);
    RSPR_scatter_kernel<<<scatterBlocks, 256, 0, stream>>>(H, cnt, src, dst, etype, Y2);

    // pool + classify (ReLU fused into pool read)
    RSPR_zero_kernel<<<(NGRAPHS * HID + NGRAPHS + ZB - 1) / ZB, ZB, 0, stream>>>(
        pooled, NGRAPHS * HID + NGRAPHS);
    RSPR_pool_kernel<<<poolBlocks, 256, 0, stream>>>(Y2, batch, pooled, pcnt);
    RSPR_final_kernel<<<(NGRAPHS * CLASSES + ZB - 1) / ZB, ZB, 0, stream>>>(
        pooled, pcnt, Wlin, blin, out);
}